// TCNModel_7928509628486
// MI455X (gfx1250) — compile-verified
//
#include <hip/hip_runtime.h>

typedef __attribute__((ext_vector_type(2))) float v2f;
typedef __attribute__((ext_vector_type(8))) float v8f;

#define HID 512
#define CIN 256
#define NT  (HID / 16)   // 32 N-tiles, all GEMMs have N=512

// ---------------------------------------------------------------------------
// Weight transpose+pack: w (O, C, Kc) row-major  ->  bt paired layout
//   j = k*C + c  (k-major concat matches A rows = [x[t-(K-1)], ..., x[t]])
//   bt[(j/2)*O*2 + o*2 + (j&1)] = w[o][c][k]
// so a WMMA B fragment lane loads its 2 elements (j, j+1) as one b64.
// ---------------------------------------------------------------------------
__global__ void tcn_transpose_pack(const float* __restrict__ w,
                                   float* __restrict__ bt,
                                   int O, int C, int Kc) {
  int tid = blockIdx.x * blockDim.x + threadIdx.x;
  int total = O * C * Kc;
  if (tid >= total) return;
  int o = tid % O;
  int j = tid / O;
  int c = j % C;
  int k = j / C;
  float v = w[((size_t)o * C + c) * Kc + k];
  bt[((size_t)(j >> 1) * O + o) * 2 + (j & 1)] = v;
}

// ---------------------------------------------------------------------------
// Core wave-level fp32 WMMA loop: 16x16 tile, K-step 4.
// A fragment:  lane(li,kh) vgpr v -> A[m = mBase+li][K = kk + 2*kh + v]
// B fragment:  lane(li,kh) vgpr v -> B[K = kk + 2*kh + v][n = nBase+li]
//   (paired bt layout: pair index j2 = kk/2 + kh, elements contiguous)
// ---------------------------------------------------------------------------
__device__ __forceinline__ v8f tcn_wmma_loop(const float* __restrict__ aPtr,
                                             const float* __restrict__ btP,
                                             int nCol, int Kdim, int kh, v8f acc) {
#pragma unroll 4
  for (int kk = 0; kk < Kdim; kk += 4) {
    v2f a = *(const v2f*)(aPtr + kk + 2 * kh);
    v2f b = *(const v2f*)(btP + ((size_t)((kk >> 1) + kh) * HID + nCol) * 2);
    acc = __builtin_amdgcn_wmma_f32_16x16x4_f32(false, a, false, b,
                                                (short)0, acc, false, false);
  }
  return acc;
}

// ---------------------------------------------------------------------------
// Stage 1: mid[(b*11+ti)][o] = relu(conv1_0(x) + b1_0) at t = 1013+ti, ti=0..10
// A row (b,ti) = x rows (b*1024 + 1011+ti .. 1013+ti) = 768 contiguous floats.
// M = 704, N = 512, K = 768.
// ---------------------------------------------------------------------------
__global__ void tcn_k_mid(const float* __restrict__ x,
                          const float* __restrict__ bt,
                          const float* __restrict__ bias,
                          float* __restrict__ mid) {
  int wave = (blockIdx.x * blockDim.x + threadIdx.x) >> 5;
  int lane = threadIdx.x & 31;
  if (wave >= (704 / 16) * NT) return;          // wave-uniform
  int mt = wave / NT, nt = wave % NT;
  int kh = lane >> 4, li = lane & 15;
  int m = mt * 16 + li;
  int b = m / 11, ti = m % 11;
  const float* aPtr = x + (size_t)(b * 1024 + 1011 + ti) * CIN;
  int nCol = nt * 16 + li;
  v8f acc = {0.f, 0.f, 0.f, 0.f, 0.f, 0.f, 0.f, 0.f};
  acc = tcn_wmma_loop(aPtr, bt, nCol, 3 * CIN, kh, acc);
  float bs = bias[nCol];
#pragma unroll
  for (int v = 0; v < 8; ++v) {
    int mr = mt * 16 + v + 8 * kh;
    float r = acc[v] + bs;
    mid[(size_t)mr * HID + nCol] = r > 0.f ? r : 0.f;
  }
}

// ---------------------------------------------------------------------------
// Stage 2: h1 row m = b*5+tj (t = 1015+2*tj, tj=0..4)
//   out = relu(conv2_0(mid[t-2..t]) + b2_0)          K = 1536 (mid rows 2tj..2tj+2)
//   res = down_w0 . x[t] + down_b0                   K = 256
//   h1  = relu(out + res)
// M = 320, N = 512.
// ---------------------------------------------------------------------------
__global__ void tcn_k_h1(const float* __restrict__ mid,
                         const float* __restrict__ bt2,
                         const float* __restrict__ b2,
                         const float* __restrict__ x,
                         const float* __restrict__ btd,
                         const float* __restrict__ bd,
                         float* __restrict__ h1) {
  int wave = (blockIdx.x * blockDim.x + threadIdx.x) >> 5;
  int lane = threadIdx.x & 31;
  if (wave >= (320 / 16) * NT) return;
  int mt = wave / NT, nt = wave % NT;
  int kh = lane >> 4, li = lane & 15;
  int m = mt * 16 + li;
  int b = m / 5, tj = m % 5;
  int nCol = nt * 16 + li;
  v8f z = {0.f, 0.f, 0.f, 0.f, 0.f, 0.f, 0.f, 0.f};
  const float* aPtr1 = mid + (size_t)(b * 11 + 2 * tj) * HID;   // 3 contiguous rows
  v8f acc1 = tcn_wmma_loop(aPtr1, bt2, nCol, 3 * HID, kh, z);
  const float* aPtr2 = x + (size_t)(b * 1024 + 1015 + 2 * tj) * CIN;
  v8f acc2 = tcn_wmma_loop(aPtr2, btd, nCol, CIN, kh, z);
  float bias2 = b2[nCol], biasd = bd[nCol];
#pragma unroll
  for (int v = 0; v < 8; ++v) {
    int mr = mt * 16 + v + 8 * kh;           // == storage row b*5+tj
    float o = acc1[v] + bias2;
    o = o > 0.f ? o : 0.f;
    float r = o + acc2[v] + biasd;
    h1[(size_t)mr * HID + nCol] = r > 0.f ? r : 0.f;
  }
}

// ---------------------------------------------------------------------------
// Stage 3: out1 row m = b*3+tc (t = 1019+2*tc): relu(conv1_1(h1) + b1_1)
// A row = h1 rows (b*5+tc .. b*5+tc+2) = 1536 contiguous (dilation-2 windows).
// M = 192, N = 512, K = 1536.
// ---------------------------------------------------------------------------
__global__ void tcn_k_out1(const float* __restrict__ h1,
                           const float* __restrict__ bt,
                           const float* __restrict__ bias,
                           float* __restrict__ out1) {
  int wave = (blockIdx.x * blockDim.x + threadIdx.x) >> 5;
  int lane = threadIdx.x & 31;
  if (wave >= (192 / 16) * NT) return;
  int mt = wave / NT, nt = wave % NT;
  int kh = lane >> 4, li = lane & 15;
  int m = mt * 16 + li;
  int b = m / 3, tc = m % 3;
  const float* aPtr = h1 + (size_t)(b * 5 + tc) * HID;
  int nCol = nt * 16 + li;
  v8f acc = {0.f, 0.f, 0.f, 0.f, 0.f, 0.f, 0.f, 0.f};
  acc = tcn_wmma_loop(aPtr, bt, nCol, 3 * HID, kh, acc);
  float bs = bias[nCol];
#pragma unroll
  for (int v = 0; v < 8; ++v) {
    int mr = mt * 16 + v + 8 * kh;
    float r = acc[v] + bs;
    out1[(size_t)mr * HID + nCol] = r > 0.f ? r : 0.f;
  }
}

// ---------------------------------------------------------------------------
// Stage 4: h2[b][o] = relu(relu(conv2_1(out1) + b2_1) + h1[t=1023])
// A row b = out1 rows (b*3 .. b*3+2) = 1536 contiguous. M = 64, N = 512.
// ---------------------------------------------------------------------------
__global__ void tcn_k_h2(const float* __restrict__ out1,
                         const float* __restrict__ bt,
                         const float* __restrict__ bias,
                         const float* __restrict__ h1,
                         float* __restrict__ h2) {
  int wave = (blockIdx.x * blockDim.x + threadIdx.x) >> 5;
  int lane = threadIdx.x & 31;
  if (wave >= (64 / 16) * NT) return;
  int mt = wave / NT, nt = wave % NT;
  int kh = lane >> 4, li = lane & 15;
  int m = mt * 16 + li;
  const float* aPtr = out1 + (size_t)m * (3 * HID);
  int nCol = nt * 16 + li;
  v8f acc = {0.f, 0.f, 0.f, 0.f, 0.f, 0.f, 0.f, 0.f};
  acc = tcn_wmma_loop(aPtr, bt, nCol, 3 * HID, kh, acc);
  float bs = bias[nCol];
#pragma unroll
  for (int v = 0; v < 8; ++v) {
    int mr = mt * 16 + v + 8 * kh;           // batch index
    float o = acc[v] + bs;
    o = o > 0.f ? o : 0.f;
    float r = o + h1[(size_t)(mr * 5 + 4) * HID + nCol];
    h2[(size_t)mr * HID + nCol] = r > 0.f ? r : 0.f;
  }
}

// ---------------------------------------------------------------------------
// Stage 5: out[b][o] = h2[b] . fc_w[o] + fc_b[o]   (64 x 36, K=512; tiny)
// ---------------------------------------------------------------------------
__global__ void tcn_k_fc(const float* __restrict__ h2,
                         const float* __restrict__ fw,
                         const float* __restrict__ fb,
                         float* __restrict__ out) {
  int id = blockIdx.x * blockDim.x + threadIdx.x;
  if (id >= 64 * 36) return;
  int b = id / 36, o = id % 36;
  float s = fb[o];
  const float* hrow = h2 + (size_t)b * HID;
  const float* wrow = fw + (size_t)o * HID;
#pragma unroll 4
  for (int c = 0; c < HID; ++c) s += hrow[c] * wrow[c];
  out[id] = s;
}

extern "C" void kernel_launch(void* const* d_in, const int* in_sizes, int n_in,
                              void* d_out, int out_size, void* d_ws, size_t ws_size,
                              hipStream_t stream) {
  const float* x       = (const float*)d_in[0];
  const float* w1_0    = (const float*)d_in[4];
  const float* b1_0    = (const float*)d_in[5];
  const float* w2_0    = (const float*)d_in[6];
  const float* b2_0    = (const float*)d_in[7];
  const float* dw0     = (const float*)d_in[8];
  const float* db0     = (const float*)d_in[9];
  const float* w1_1    = (const float*)d_in[10];
  const float* b1_1    = (const float*)d_in[11];
  const float* w2_1    = (const float*)d_in[12];
  const float* b2_1    = (const float*)d_in[13];
  const float* fc_w    = (const float*)d_in[14];
  const float* fc_b    = (const float*)d_in[15];
  float* out = (float*)d_out;

  // Workspace layout (floats); total 6912*512 = ~13.5 MiB
  float* ws    = (float*)d_ws;
  float* bt1_0 = ws;                       // 768*512
  float* bt2_0 = bt1_0 + 768 * 512;        // 1536*512
  float* btd   = bt2_0 + 1536 * 512;       // 256*512
  float* bt1_1 = btd + 256 * 512;          // 1536*512
  float* bt2_1 = bt1_1 + 1536 * 512;       // 1536*512
  float* mid   = bt2_1 + 1536 * 512;       // 704*512
  float* h1    = mid + 704 * 512;          // 320*512
  float* out1  = h1 + 320 * 512;           // 192*512
  float* h2    = out1 + 192 * 512;         // 64*512

  // Pack weights into WMMA-B paired layout
  tcn_transpose_pack<<<(512 * 256 * 3 + 255) / 256, 256, 0, stream>>>(w1_0, bt1_0, 512, 256, 3);
  tcn_transpose_pack<<<(512 * 512 * 3 + 255) / 256, 256, 0, stream>>>(w2_0, bt2_0, 512, 512, 3);
  tcn_transpose_pack<<<(512 * 256 * 1 + 255) / 256, 256, 0, stream>>>(dw0,  btd,   512, 256, 1);
  tcn_transpose_pack<<<(512 * 512 * 3 + 255) / 256, 256, 0, stream>>>(w1_1, bt1_1, 512, 512, 3);
  tcn_transpose_pack<<<(512 * 512 * 3 + 255) / 256, 256, 0, stream>>>(w2_1, bt2_1, 512, 512, 3);

  // GEMM stages: 1 wave per 16x16 tile, 4 waves (128 threads) per block
  tcn_k_mid <<<(44 * NT) / 4, 128, 0, stream>>>(x, bt1_0, b1_0, mid);
  tcn_k_h1  <<<(20 * NT) / 4, 128, 0, stream>>>(mid, bt2_0, b2_0, x, btd, db0, h1);
  tcn_k_out1<<<(12 * NT) / 4, 128, 0, stream>>>(h1, bt1_1, b1_1, out1);
  tcn_k_h2  <<<( 4 * NT) / 4, 128, 0, stream>>>(out1, bt2_1, b2_1, h1, h2);
  tcn_k_fc  <<<(64 * 36 + 255) / 256, 256, 0, stream>>>(h2, fc_w, fc_b, out);
}